// ActorNetwork_77043123355907
// MI455X (gfx1250) — compile-verified
//
#include <hip/hip_runtime.h>

typedef __attribute__((ext_vector_type(2))) float v2f;
typedef __attribute__((ext_vector_type(8))) float v8f;

#define VOCAB 87429
#define EDIM  51
#define CCH   39
#define HLEN  8192
#define NCOLS 38   // sum of kernel sizes 1+2+5+10+20
#define NPAD  48   // 3 WMMA N-tiles of 16
#define KPADC 52   // per-channel K (51) padded to multiple of 4 -> 13 K=4 steps

// ---------------------------------------------------------------------------
// Kernel 1: pack the 5 conv-weight tensors into a dense, zero-padded B matrix
//   Bmat[(c*KPADC + e)*NPAD + m] = w_i[c, j, e]  with m -> (branch i, offset j)
// Column layout: branch0 (k=1): m=0 | branch1 (k=2): m=1..2 | branch2 (k=5):
// m=3..7 | branch3 (k=10): m=8..17 | branch4 (k=20): m=18..37. Also zeroes the
// 5 feature-max accumulators (bits 0 == 0.0f).
// ---------------------------------------------------------------------------
__global__ void pack_b_kernel(const float* __restrict__ w0, const float* __restrict__ w1,
                              const float* __restrict__ w2, const float* __restrict__ w3,
                              const float* __restrict__ w4, float* __restrict__ Bmat,
                              float* __restrict__ feats) {
  int idx = blockIdx.x * blockDim.x + threadIdx.x;
  const int total = CCH * KPADC * NPAD;
  if (idx < 5) feats[idx] = 0.0f;
  if (idx >= total) return;
  int m  = idx % NPAD;
  int ce = idx / NPAD;
  int e  = ce % KPADC;
  int c  = ce / KPADC;
  float v = 0.0f;
  if (m < NCOLS && e < EDIM) {
    int i, j, kk;
    if      (m < 1)  { i = 0; j = m;      kk = 1;  }
    else if (m < 3)  { i = 1; j = m - 1;  kk = 2;  }
    else if (m < 8)  { i = 2; j = m - 3;  kk = 5;  }
    else if (m < 18) { i = 3; j = m - 8;  kk = 10; }
    else             { i = 4; j = m - 18; kk = 20; }
    const float* wp = (i == 0) ? w0 : (i == 1) ? w1 : (i == 2) ? w2 : (i == 3) ? w3 : w4;
    v = wp[(c * kk + j) * EDIM + e];
  }
  Bmat[idx] = v;
}

// ---------------------------------------------------------------------------
// Kernel 2: fused gather+GEMM with fp32 WMMA.
//   Q[h, m] = sum_{c,e} emb[tok[c,h], e] * Bmat[c*KPADC+e, m]
// One 16-row M tile per wave; 3 N tiles of 16 in registers; K loop over
// 39 channels x 13 steps of K=4 using V_WMMA_F32_16X16X4_F32.
// A layout (16x4 f32): lanes 0-15 hold rows M=lane, K=0/1 in v0/v1;
// lanes 16-31 hold rows M=lane-16, K=2/3. B (4x16) mirrors with K striped.
// ---------------------------------------------------------------------------
__global__ __launch_bounds__(128) void gemm_kernel(const int* __restrict__ tokens,
                                                   const float* __restrict__ emb,
                                                   const float* __restrict__ Bmat,
                                                   float* __restrict__ Q) {
  const int wave = threadIdx.x >> 5;
  const int lane = threadIdx.x & 31;
  const int half = lane >> 4;     // 0: K pair {0,1}, 1: K pair {2,3}
  const int l16  = lane & 15;
  const int tile = blockIdx.x * 4 + wave;   // 512 tiles total
  const int row  = tile * 16 + l16;

  v8f acc0 = {0.f, 0.f, 0.f, 0.f, 0.f, 0.f, 0.f, 0.f};
  v8f acc1 = acc0;
  v8f acc2 = acc0;

  for (int c = 0; c < CCH; ++c) {
    const int tok = tokens[c * HLEN + row];
    const float* erow = emb + (long)tok * EDIM;
#pragma unroll
    for (int s = 0; s < 13; ++s) {
      const int kb = s * 4 + half * 2;   // this lane's K pair base within channel
      v2f a;
      a.x = (kb     < EDIM) ? erow[kb]     : 0.0f;
      a.y = (kb + 1 < EDIM) ? erow[kb + 1] : 0.0f;
      const float* bp = Bmat + (c * KPADC + kb) * NPAD + l16;  // KPADC=52 keeps kb+1<=51 in-bounds (zero pad)
      v2f b0; b0.x = bp[0];  b0.y = bp[NPAD];
      v2f b1; b1.x = bp[16]; b1.y = bp[16 + NPAD];
      v2f b2; b2.x = bp[32]; b2.y = bp[32 + NPAD];
      acc0 = __builtin_amdgcn_wmma_f32_16x16x4_f32(false, a, false, b0, (short)0, acc0, false, false);
      acc1 = __builtin_amdgcn_wmma_f32_16x16x4_f32(false, a, false, b1, (short)0, acc1, false, false);
      acc2 = __builtin_amdgcn_wmma_f32_16x16x4_f32(false, a, false, b2, (short)0, acc2, false, false);
    }
  }

  // C/D layout: VGPR r, lanes 0-15 -> row M=r, lanes 16-31 -> row M=r+8
#pragma unroll
  for (int r = 0; r < 8; ++r) {
    const int orow = tile * 16 + half * 8 + r;
    Q[orow * NPAD +      l16] = acc0[r];
    Q[orow * NPAD + 16 + l16] = acc1[r];
    Q[orow * NPAD + 32 + l16] = acc2[r];
  }
}

// ---------------------------------------------------------------------------
// Kernel 3: per-branch shifted-diagonal sum + global max(relu(.)).
//   y_i[h] = b_i + sum_j Q[h+j, start_i + j],  h in [0, HLEN-k_i]
// Wave-shuffle max then atomicMax on float bits (all candidates >= 0).
// ---------------------------------------------------------------------------
__global__ void reduce_kernel(const float* __restrict__ Q,
                              const float* __restrict__ b0, const float* __restrict__ b1,
                              const float* __restrict__ b2, const float* __restrict__ b3,
                              const float* __restrict__ b4,
                              unsigned int* __restrict__ feats) {
  const int h = blockIdx.x * blockDim.x + threadIdx.x;
  const int starts[5] = {0, 1, 3, 8, 18};
  const int kks[5]    = {1, 2, 5, 10, 20};
  const float bias[5] = {b0[0], b1[0], b2[0], b3[0], b4[0]};
#pragma unroll
  for (int i = 0; i < 5; ++i) {
    float v = 0.0f;
    if (h <= HLEN - kks[i]) {
      float y = bias[i];
      for (int j = 0; j < kks[i]; ++j)
        y += Q[(h + j) * NPAD + starts[i] + j];
      v = fmaxf(y, 0.0f);
    }
    for (int off = 16; off > 0; off >>= 1)
      v = fmaxf(v, __shfl_xor(v, off, 32));
    if ((threadIdx.x & 31) == 0)
      atomicMax(&feats[i], __float_as_uint(v));
  }
}

// ---------------------------------------------------------------------------
// Kernel 4: 8x5 linear + relu + softmax (trivial scalar tail).
// ---------------------------------------------------------------------------
__global__ void head_kernel(const unsigned int* __restrict__ feats,
                            const float* __restrict__ lin_w,
                            const float* __restrict__ lin_b,
                            float* __restrict__ out) {
  if (threadIdx.x == 0) {
    float f[5];
    for (int i = 0; i < 5; ++i) f[i] = __uint_as_float(feats[i]);
    float lg[8];
    float mx = -1e30f;
    for (int q = 0; q < 8; ++q) {
      float s = lin_b[q];
      for (int i = 0; i < 5; ++i) s += lin_w[q * 5 + i] * f[i];
      s = fmaxf(s, 0.0f);
      lg[q] = s;
      mx = fmaxf(mx, s);
    }
    float sum = 0.0f;
    for (int q = 0; q < 8; ++q) { lg[q] = expf(lg[q] - mx); sum += lg[q]; }
    for (int q = 0; q < 8; ++q) out[q] = lg[q] / sum;
  }
}

extern "C" void kernel_launch(void* const* d_in, const int* in_sizes, int n_in,
                              void* d_out, int out_size, void* d_ws, size_t ws_size,
                              hipStream_t stream) {
  (void)in_sizes; (void)n_in; (void)out_size; (void)ws_size;
  const int*   tokens = (const int*)  d_in[0];
  const float* emb    = (const float*)d_in[1];
  const float* lin_w  = (const float*)d_in[2];
  const float* lin_b  = (const float*)d_in[3];
  const float* w0 = (const float*)d_in[4];  const float* b0 = (const float*)d_in[5];
  const float* w1 = (const float*)d_in[6];  const float* b1 = (const float*)d_in[7];
  const float* w2 = (const float*)d_in[8];  const float* b2 = (const float*)d_in[9];
  const float* w3 = (const float*)d_in[10]; const float* b3 = (const float*)d_in[11];
  const float* w4 = (const float*)d_in[12]; const float* b4 = (const float*)d_in[13];

  float* ws   = (float*)d_ws;
  float* Bmat = ws;                                   // 39*52*48 = 97,344 floats
  float* Q    = Bmat + CCH * KPADC * NPAD;            // 8192*48  = 393,216 floats
  unsigned int* feats = (unsigned int*)(Q + HLEN * NPAD);  // 5 words; ~2 MB total

  const int totalB = CCH * KPADC * NPAD;
  pack_b_kernel<<<(totalB + 255) / 256, 256, 0, stream>>>(w0, w1, w2, w3, w4, Bmat, (float*)feats);
  gemm_kernel<<<HLEN / 16 / 4, 128, 0, stream>>>(tokens, emb, Bmat, Q);
  reduce_kernel<<<HLEN / 256, 256, 0, stream>>>(Q, b0, b1, b2, b3, b4, feats);
  head_kernel<<<1, 32, 0, stream>>>(feats, lin_w, lin_b, (float*)d_out);
}